// VTrace_62646392979724
// MI455X (gfx1250) — compile-verified
//
#include <hip/hip_runtime.h>
#include <hip/hip_bf16.h>

// V-trace backward scan, MI455X (gfx1250).
// Memory-bound streaming kernel: ~805 MB moved, 23.3 TB/s HBM -> ~35 us floor.
// One thread per batch column (65536 cols = 2048 wave32s), backward over s.
// CDNA5 paths: global_prefetch_b8 (L2 priming of the backward stream),
// TH_*_NT cache hints (every word touched exactly once vs 192MB L2),
// v_exp_f32 TRANS op (co-executes with VALU), and GVS/SADDR addressing:
// uniform scalar base pointers walk the rows (SALU induction) while the
// per-lane VGPR offset (col*4) is loop-invariant -> address math stays off
// the VALU issue port.

#define VT_GAMMA   0.99f
#define VT_PF_DIST 16   // prefetch distance in rows: 16 * 4 arrays * 256KB = 16MB L2 window

__global__ __launch_bounds__(256) void vtrace_scan_kernel(
    const float* __restrict__ V,       // (S+1, B) target_value
    const float* __restrict__ rew,     // (S,   B) rewards
    const float* __restrict__ tlp,     // (S,   B) target_log_policy
    const float* __restrict__ blp,     // (S,   B) behaviour_log_policy
    float* __restrict__ out_vt,        // (S+1, B) vtrace
    float* __restrict__ out_rho,       // (S,   B) clipped_rhos
    int S, int B)
{
    const int b = blockIdx.x * blockDim.x + threadIdx.x;
    if (b >= B) return;
    const int    col = b;          // loop-invariant per-lane offset (VGPR32)
    const size_t Bsz = (size_t)B;

    // vtrace[S] = V[S]; seed v_next with it.
    const float* pTop = V      + (size_t)S * Bsz;
    float*       oTop = out_vt + (size_t)S * Bsz;
    float v_next = __builtin_nontemporal_load(pTop + col);
    __builtin_nontemporal_store(v_next, oTop + col);

    // Uniform (SGPR) row pointers, walking backward from row S-1.
    const size_t r0 = (size_t)(S - 1) * Bsz;
    const float* pT  = tlp     + r0;
    const float* pBl = blp     + r0;
    const float* pR  = rew     + r0;
    const float* pV  = V       + r0;
    float*       pOv = out_vt  + r0;
    float*       pOr = out_rho + r0;
    // Prefetch pointers VT_PF_DIST rows ahead of the consume point.
    const size_t rp = (size_t)VT_PF_DIST * Bsz;
    const float* qT  = pT  - rp;
    const float* qBl = pBl - rp;
    const float* qR  = pR  - rp;
    const float* qV  = pV  - rp;

    float a = 0.0f;
    int s = S - 1;

    // Body: a_s = rho_s*(r_s + g*V[s+1] - V[s]) + g*rho_s*a_{s+1}
    // RHO == CIS == 1.0 -> clipped_rho and cis are the same value.
    // rho = min(1, exp(x)) == exp(min(0, x))  (exp monotone) -> 1 TRANS op.
#define VT_BODY()                                                           \
    {                                                                       \
        const float tl = __builtin_nontemporal_load(pT  + col);             \
        const float bl = __builtin_nontemporal_load(pBl + col);             \
        const float rw = __builtin_nontemporal_load(pR  + col);             \
        const float vc = __builtin_nontemporal_load(pV  + col);             \
        const float rho = __expf(fminf(0.0f, tl - bl));                     \
        const float delta = rho * (fmaf(VT_GAMMA, v_next, rw) - vc);        \
        a = fmaf(VT_GAMMA * rho, a, delta);                                 \
        __builtin_nontemporal_store(vc + a, pOv + col);                     \
        __builtin_nontemporal_store(rho,    pOr + col);                     \
        v_next = vc;                                                        \
        pT -= Bsz; pBl -= Bsz; pR -= Bsz; pV -= Bsz;                        \
        pOv -= Bsz; pOr -= Bsz;                                             \
    }

    // Main loop: unconditional software prefetch (-> global_prefetch_b8).
    // Backward walk through memory; prime L2 VT_PF_DIST rows ahead.
    #pragma unroll 8
    for (; s >= VT_PF_DIST; --s) {
        __builtin_prefetch(qT  + col, 0, 0);
        __builtin_prefetch(qBl + col, 0, 0);
        __builtin_prefetch(qR  + col, 0, 0);
        __builtin_prefetch(qV  + col, 0, 0);
        qT -= Bsz; qBl -= Bsz; qR -= Bsz; qV -= Bsz;
        VT_BODY()
    }

    // Tail: first VT_PF_DIST rows, already prefetched by the main loop.
    #pragma unroll 4
    for (; s >= 0; --s) {
        VT_BODY()
    }
#undef VT_BODY
}

extern "C" void kernel_launch(void* const* d_in, const int* in_sizes, int n_in,
                              void* d_out, int out_size, void* d_ws, size_t ws_size,
                              hipStream_t stream)
{
    (void)n_in; (void)d_ws; (void)ws_size; (void)out_size;

    // setup_inputs() order: target_value (S+1,B), rewards (S,B),
    //                       target_log_policy (S,B), behaviour_log_policy (S,B)
    const float* V   = (const float*)d_in[0];
    const float* rew = (const float*)d_in[1];
    const float* tlp = (const float*)d_in[2];
    const float* blp = (const float*)d_in[3];

    const int SB = in_sizes[1];          // S * B
    const int B  = in_sizes[0] - SB;     // (S+1)*B - S*B = B
    const int S  = SB / B;

    // d_out = [ vtrace (S+1,B) | clipped_rhos (S,B) ] flattened in return order.
    float* out_vt  = (float*)d_out;
    float* out_rho = out_vt + (size_t)(S + 1) * (size_t)B;

    const int threads = 256;
    const int blocks  = (B + threads - 1) / threads;
    vtrace_scan_kernel<<<blocks, threads, 0, stream>>>(V, rew, tlp, blp,
                                                       out_vt, out_rho, S, B);
}